// GATLayer_23768349016121
// MI455X (gfx1250) — compile-verified
//
#include <hip/hip_runtime.h>

#define ALPHA 0.2f

typedef float v2f __attribute__((ext_vector_type(2)));
typedef float v8f __attribute__((ext_vector_type(8)));

constexpr int Bn    = 8;
constexpr int Nn    = 2048;
constexpr int F_IN  = 128;
constexpr int F_OUT = 64;
constexpr int ROWS  = Bn * Nn;          // 16384 flattened rows of h

// ---------------------------------------------------------------------------
// Kernel 1: Wh = h @ W via V_WMMA_F32_16X16X4_F32.
// Grid: 1024 blocks (one 16-row M tile each), 128 threads = 4 waves,
// wave w owns output columns [16w, 16w+16).
// ---------------------------------------------------------------------------
__global__ __launch_bounds__(128) void k_wh_wmma(const float* __restrict__ h,
                                                 const float* __restrict__ W,
                                                 float* __restrict__ Wh) {
    __shared__ float Atile[16 * F_IN];              // 8 KB
    const int m0  = blockIdx.x * 16;
    const int tid = threadIdx.x;

    // Cooperative load of the 16x128 f32 A tile (512 float4s over 128 threads).
    const float4* src = reinterpret_cast<const float4*>(h + m0 * F_IN);
    float4*       dst = reinterpret_cast<float4*>(Atile);
#pragma unroll
    for (int q = 0; q < 4; ++q) dst[tid + 128 * q] = src[tid + 128 * q];
    __syncthreads();

    const int wave = tid >> 5;                      // tileN index 0..3
    const int lane = tid & 31;
    const int half = lane >> 4;                     // 0: lanes 0-15, 1: lanes 16-31
    const int m    = lane & 15;                     // row-in-tile (A), col-in-tile (B/D)
    const int col  = wave * 16 + m;                 // output column 0..63

    v8f c = {0.f, 0.f, 0.f, 0.f, 0.f, 0.f, 0.f, 0.f};

#pragma unroll
    for (int kk = 0; kk < F_IN / 4; ++kk) {         // 32 WMMA steps, K=4 each
        const int k0 = kk * 4 + 2 * half;           // ISA: v0 holds K={0,2}, v1 K={1,3}
        v2f a, b;
        a.x = Atile[m * F_IN + k0];
        a.y = Atile[m * F_IN + k0 + 1];
        b.x = W[(k0)     * F_OUT + col];
        b.y = W[(k0 + 1) * F_OUT + col];
        c = __builtin_amdgcn_wmma_f32_16x16x4_f32(false, a, false, b,
                                                  (short)0, c, false, false);
    }

    // D layout: VGPR r -> M = r + 8*half, N = lane&15.
#pragma unroll
    for (int r = 0; r < 8; ++r) {
        const int row = m0 + r + 8 * half;
        Wh[row * F_OUT + col] = c[r];
    }
}

// ---------------------------------------------------------------------------
// Kernel 2: per-row attention logit halves  s1 = Wh@a[:64], s2 = Wh@a[64:].
// ---------------------------------------------------------------------------
__global__ __launch_bounds__(256) void k_scores(const float* __restrict__ Wh,
                                                const float* __restrict__ a,
                                                float* __restrict__ s1,
                                                float* __restrict__ s2) {
    const int row = blockIdx.x * 256 + threadIdx.x;       // 16384 rows
    const float4* wr = reinterpret_cast<const float4*>(Wh + row * F_OUT);
    const float4* a1 = reinterpret_cast<const float4*>(a);
    const float4* a2 = reinterpret_cast<const float4*>(a + F_OUT);
    float acc1 = 0.f, acc2 = 0.f;
#pragma unroll
    for (int q = 0; q < F_OUT / 4; ++q) {
        const float4 w4 = wr[q];
        const float4 x  = a1[q];
        const float4 y  = a2[q];
        acc1 += w4.x * x.x + w4.y * x.y + w4.z * x.z + w4.w * x.w;
        acc2 += w4.x * y.x + w4.y * y.y + w4.z * y.z + w4.w * y.w;
    }
    s1[row] = acc1;
    s2[row] = acc2;
}

// ---------------------------------------------------------------------------
// Kernel 3: softmax row denominators  rZ[b,i] = 1 / sum_j exp(lrelu(s1_i+s2_j)).
// grid (8 i-chunks, 8 batches) x 256 threads; s2[b,:] staged in LDS.
// ---------------------------------------------------------------------------
__global__ __launch_bounds__(256) void k_rowsum(const float* __restrict__ s1,
                                                const float* __restrict__ s2,
                                                float* __restrict__ rZ) {
    __shared__ float sj[Nn];                        // 8 KB
    const int b = blockIdx.y;
    for (int q = threadIdx.x; q < Nn; q += 256) sj[q] = s2[b * Nn + q];
    __syncthreads();

    const int i  = blockIdx.x * 256 + threadIdx.x;
    const float si = s1[b * Nn + i];
    float z = 0.f;
#pragma unroll 8
    for (int j = 0; j < Nn; ++j) {
        float t = si + sj[j];
        t = (t > 0.f) ? t : ALPHA * t;
        z += __expf(t);
    }
    rZ[b * Nn + i] = 1.0f / z;
}

// ---------------------------------------------------------------------------
// Kernel 4: attention column weights  c[b,j] = sum_i exp(lrelu(s1_i+s2_j))*rZ_i.
// ---------------------------------------------------------------------------
__global__ __launch_bounds__(256) void k_colsum(const float* __restrict__ s1,
                                                const float* __restrict__ s2,
                                                const float* __restrict__ rZ,
                                                float* __restrict__ cw) {
    __shared__ float si[Nn];                        // 8 KB
    __shared__ float rz[Nn];                        // 8 KB
    const int b = blockIdx.y;
    for (int q = threadIdx.x; q < Nn; q += 256) {
        si[q] = s1[b * Nn + q];
        rz[q] = rZ[b * Nn + q];
    }
    __syncthreads();

    const int j  = blockIdx.x * 256 + threadIdx.x;
    const float sjv = s2[b * Nn + j];
    float acc = 0.f;
#pragma unroll 8
    for (int i = 0; i < Nn; ++i) {
        float t = si[i] + sjv;
        t = (t > 0.f) ? t : ALPHA * t;
        acc += __expf(t) * rz[i];
    }
    cw[b * Nn + j] = acc;
}

// ---------------------------------------------------------------------------
// Kernel 5: out[b,f] = (1/N) * sum_j cw[b,j] * Wh[b,j,f].
// One block per batch, 4 j-chunks x 64 f-lanes, LDS reduction.
// ---------------------------------------------------------------------------
__global__ __launch_bounds__(256) void k_out(const float* __restrict__ Wh,
                                             const float* __restrict__ cw,
                                             float* __restrict__ out) {
    __shared__ float red[256];
    const int b     = blockIdx.x;
    const int f     = threadIdx.x & 63;
    const int chunk = threadIdx.x >> 6;
    float acc = 0.f;
    for (int j = chunk; j < Nn; j += 4)
        acc += cw[b * Nn + j] * Wh[(b * Nn + j) * F_OUT + f];
    red[threadIdx.x] = acc;
    __syncthreads();
    if (threadIdx.x < 64) {
        const float s = red[f] + red[64 + f] + red[128 + f] + red[192 + f];
        out[b * F_OUT + f] = s * (1.0f / Nn);
    }
}

// ---------------------------------------------------------------------------
extern "C" void kernel_launch(void* const* d_in, const int* in_sizes, int n_in,
                              void* d_out, int out_size, void* d_ws, size_t ws_size,
                              hipStream_t stream) {
    (void)in_sizes; (void)n_in; (void)out_size; (void)ws_size;
    const float* h = (const float*)d_in[0];   // [8,2048,128]
    const float* W = (const float*)d_in[1];   // [128,64]
    const float* a = (const float*)d_in[2];   // [128]
    float* out = (float*)d_out;               // [8,64]

    float* ws  = (float*)d_ws;                // scratch layout (floats):
    float* Wh  = ws;                          //   16384*64  = 4 MB
    float* s1  = Wh + ROWS * F_OUT;           //   16384
    float* s2  = s1 + ROWS;                   //   16384
    float* rZ  = s2 + ROWS;                   //   16384
    float* cw  = rZ + ROWS;                   //   16384   (total ~4.25 MB)

    k_wh_wmma<<<ROWS / 16, 128, 0, stream>>>(h, W, Wh);
    k_scores <<<ROWS / 256, 256, 0, stream>>>(Wh, a, s1, s2);
    dim3 g(Nn / 256, Bn);
    k_rowsum <<<g, 256, 0, stream>>>(s1, s2, rZ);
    k_colsum <<<g, 256, 0, stream>>>(s1, s2, rZ, cw);
    k_out    <<<Bn, 256, 0, stream>>>(Wh, cw, out);
}